// BlockDecomposition_5265629905642
// MI455X (gfx1250) — compile-verified
//
#include <hip/hip_runtime.h>
#include <hip/hip_bf16.h>

typedef __attribute__((ext_vector_type(2))) float v2f;
typedef __attribute__((ext_vector_type(8))) float v8f;

#define DIM   128
#define NBLK  8
#define BS    16
#define NREL  8

// ---------------------------------------------------------------------------
// Kernel 1: self-relation block-diagonal transform with V_WMMA_F32_16X16X4_F32
//   out[n, b*16+j] = keep[n] ? sum_i x[n, b*16+i] * blocks[8][b][i][j] : 0
// One wave per (16-node tile, block b). K=16 done as 4 chained K=4 WMMAs.
// Fully writes d_out (harness poisons it); edge kernel then atomically adds.
// ---------------------------------------------------------------------------
__global__ __launch_bounds__(256) void rgcn_self_wmma_kernel(
    const float* __restrict__ x, const float* __restrict__ blocks,
    const unsigned char* __restrict__ keep, float* __restrict__ out,
    int n_nodes)
{
  const int b    = threadIdx.x >> 5;   // wave id 0..7 == feature block
  const int lane = threadIdx.x & 31;
  const int tile = blockIdx.x;
  const int hi   = lane >> 4;          // 0: lanes 0-15, 1: lanes 16-31
  const int mrow = lane & 15;

  // ---- A fragments: 16 nodes x 16 feats of block b, split into 4 K-chunks.
  // A(16x4) layout: lane = hi*16 + m, vgpr v -> k_in_chunk = v + 2*hi
  // => each lane loads a contiguous float2.
  int anode = tile * 16 + mrow;
  if (anode >= n_nodes) anode = n_nodes - 1;   // clamp (stores are guarded)
  const float* arow = x + (size_t)anode * DIM + b * BS + 2 * hi;
  v2f a[4];
#pragma unroll
  for (int c = 0; c < 4; ++c)
    a[c] = *(const v2f*)(arow + c * 4);

  // ---- B fragments: blocks[NREL][b] (16x16), chunk c covers rows 4c..4c+3.
  // B(4x16) layout mirrors A: vgpr v -> k = 4c + v + 2*hi, n = lane&15.
  const float* Bp = blocks + (size_t)(NREL * NBLK + b) * BS * BS + (lane & 15);
  v2f bf[4];
#pragma unroll
  for (int c = 0; c < 4; ++c) {
    const int k0 = c * 4 + 2 * hi;
    v2f t;
    t.x = Bp[(k0 + 0) * BS];
    t.y = Bp[(k0 + 1) * BS];
    bf[c] = t;
  }

  // ---- 4 chained f32 WMMAs (D = A*B + C), full fp32 precision.
  v8f acc = {0.f, 0.f, 0.f, 0.f, 0.f, 0.f, 0.f, 0.f};
#pragma unroll
  for (int c = 0; c < 4; ++c)
    acc = __builtin_amdgcn_wmma_f32_16x16x4_f32(
        /*neg_a=*/false, a[c], /*neg_b=*/false, bf[c],
        /*c_mod=*/(short)0, acc, /*reuse_a=*/false, /*reuse_b=*/false);

  // ---- masked store. C/D layout: vgpr i, lane -> m = i + 8*hi, n = lane&15.
  // This lane's 8 rows are nodes tile*16 + 8*hi + (0..7): their keep bytes are
  // 8 contiguous, 8B-aligned bytes -> single b64 load, branch-free cndmask.
  const int   row0 = tile * 16 + 8 * hi;             // first of this lane's rows
  const int   ncol = b * BS + (lane & 15);
  float* orow = out + (size_t)row0 * DIM + ncol;

  if (row0 + 8 <= n_nodes) {                         // full-tile fast path
    const unsigned long long kb =
        *(const unsigned long long*)(keep + row0);
#pragma unroll
    for (int i = 0; i < 8; ++i) {
      const float v = ((kb >> (8 * i)) & 0xffull) ? acc[i] : 0.0f;
      orow[(size_t)i * DIM] = v;
    }
  } else {                                           // ragged tail (unused for N%16==0)
#pragma unroll
    for (int i = 0; i < 8; ++i) {
      const int node = row0 + i;
      if (node < n_nodes)
        orow[(size_t)i * DIM] = keep[node] ? acc[i] : 0.0f;
    }
  }
}

// ---------------------------------------------------------------------------
// Kernel 2: fused per-edge transform + atomic scatter.
// One wave per edge. x (25.6 MB) and out (25.6 MB) are L2-resident (192 MB L2)
// so the 1.6M row gathers and ~205M f32 atomics stay on-chip.
//   out[t] += w * (xb[s] @ B[r]);   out[s] += w * (xb[t] @ B[r])
// Lane owns 4 output columns: block b = lane/4, cols j0..j0+3.
// r/s/t are wave-uniform: readfirstlane them into SGPRs so gathers/atomics use
// the scalar-base (SADDR) addressing form instead of per-lane 64-bit adds.
// ---------------------------------------------------------------------------
__global__ __launch_bounds__(256) void rgcn_edge_scatter_kernel(
    const float* __restrict__ x, const float* __restrict__ blocks,
    const int* __restrict__ src, const int* __restrict__ tgt,
    const int* __restrict__ etype, const float* __restrict__ w,
    float* __restrict__ out, int n_edges)
{
  const int wave = blockIdx.x * 8 + (threadIdx.x >> 5);
  if (wave >= n_edges) return;                // whole-wave uniform guard
  const int lane = threadIdx.x & 31;

  const int   r  = __builtin_amdgcn_readfirstlane(etype[wave]);
  const int   s  = __builtin_amdgcn_readfirstlane(src[wave]);
  const int   t  = __builtin_amdgcn_readfirstlane(tgt[wave]);
  const float we = w[wave];

  const int b  = lane >> 2;          // feature block 0..7
  const int j0 = (lane & 3) << 2;    // column-in-block 0,4,8,12

  const float* Bp  = blocks + (size_t)(r * NBLK + b) * BS * BS + j0;
  const float* xsp = x + (size_t)s * DIM + b * BS;
  const float* xtp = x + (size_t)t * DIM + b * BS;

  float xs[16], xt[16];
#pragma unroll
  for (int q = 0; q < 4; ++q) {
    const float4 vs = *(const float4*)(xsp + 4 * q);
    const float4 vt = *(const float4*)(xtp + 4 * q);
    xs[4 * q + 0] = vs.x; xs[4 * q + 1] = vs.y;
    xs[4 * q + 2] = vs.z; xs[4 * q + 3] = vs.w;
    xt[4 * q + 0] = vt.x; xt[4 * q + 1] = vt.y;
    xt[4 * q + 2] = vt.z; xt[4 * q + 3] = vt.w;
  }

  float4 af = {0.f, 0.f, 0.f, 0.f};  // forward message (from s, scattered to t)
  float4 ab = {0.f, 0.f, 0.f, 0.f};  // backward message (from t, scattered to s)
#pragma unroll
  for (int i = 0; i < 16; ++i) {
    const float4 Br = *(const float4*)(Bp + i * BS);  // B[r][b][i][j0..j0+3]
    af.x += xs[i] * Br.x; af.y += xs[i] * Br.y;
    af.z += xs[i] * Br.z; af.w += xs[i] * Br.w;
    ab.x += xt[i] * Br.x; ab.y += xt[i] * Br.y;
    ab.z += xt[i] * Br.z; ab.w += xt[i] * Br.w;
  }

  float* of = out + (size_t)t * DIM + b * BS + j0;
  unsafeAtomicAdd(of + 0, we * af.x);
  unsafeAtomicAdd(of + 1, we * af.y);
  unsafeAtomicAdd(of + 2, we * af.z);
  unsafeAtomicAdd(of + 3, we * af.w);

  float* ob = out + (size_t)s * DIM + b * BS + j0;
  unsafeAtomicAdd(ob + 0, we * ab.x);
  unsafeAtomicAdd(ob + 1, we * ab.y);
  unsafeAtomicAdd(ob + 2, we * ab.z);
  unsafeAtomicAdd(ob + 3, we * ab.w);
}

extern "C" void kernel_launch(void* const* d_in, const int* in_sizes, int n_in,
                              void* d_out, int out_size, void* d_ws, size_t ws_size,
                              hipStream_t stream) {
  const float*         x      = (const float*)d_in[0];
  const float*         blocks = (const float*)d_in[1];
  const unsigned char* keep   = (const unsigned char*)d_in[2]; // jnp bool -> u8
  const int*           src    = (const int*)d_in[3];
  const int*           tgt    = (const int*)d_in[4];
  const int*           et     = (const int*)d_in[5];
  const float*         w      = (const float*)d_in[6];
  float*               out    = (float*)d_out;

  const int n_nodes = in_sizes[0] / DIM;   // 50000
  const int n_edges = in_sizes[3];         // 800000

  // Kernel 1: one wave per (16-node tile, block). 256 threads = 8 waves = 8 blocks.
  const int tiles = (n_nodes + 15) / 16;
  rgcn_self_wmma_kernel<<<tiles, 256, 0, stream>>>(x, blocks, keep, out, n_nodes);

  // Kernel 2: one wave per edge, 8 waves per 256-thread block.
  const int eblocks = (n_edges + 7) / 8;
  rgcn_edge_scatter_kernel<<<eblocks, 256, 0, stream>>>(x, blocks, src, tgt, et,
                                                        w, out, n_edges);
}